// EfficientDetWithPostProcess_21320217657877
// MI455X (gfx1250) — compile-verified
//
#include <hip/hip_runtime.h>
#include <stdint.h>

// ---------------------------------------------------------------------------
// EfficientDet post-process for MI455X (gfx1250).
// Pipeline:
//  K0 zero_ws      : zero histograms + control block
//  K1 score_kernel : async-DMA classification tiles -> LDS, wave32 max/argmax,
//                    sigmoid+threshold, coarse 64K-bin histogram (bits>>16)
//  K2 scan_hist(0) : find coarse cutoff bin for top-100
//  K3 fine_hist_k  : histogram low 16 bits inside cutoff bin
//  K4 scan_hist(1) : exact 32-bit threshold bit pattern of 100th score
//  K5 compact_k    : gather ~100-150 candidates (score,idx)
//  K6 select_k     : 100 exact argmax rounds, decode+clip boxes for winners
// Workspace requirement: 4*(131088 + 2*2048 + 2*A) bytes ~= 2.2 MB.
// ---------------------------------------------------------------------------

#define NUM_CLASSES 90
#define TILE_A      64
#define TILE_CH     1440          // 64 anchors * 90 f32 * 4B / 16B chunks
#define CHUNK_LOOPS 6             // ceil(1440 / 256)
#define THREADS     256
#define MAX_DET     100
#define CAND_CAP    2048
#define SCORE_TH    0.05f
#define IMG_W       1024.0f
#define IMG_H       1024.0f

// The async-to-LDS builtin takes pointers to 16-byte GCC vectors:
//   (AS1 v4i* src, AS3 v4i* dst, imm offset, imm cpol)
typedef int v4i __attribute__((vector_size(16)));
typedef __attribute__((address_space(1))) v4i as1_v4i;
typedef __attribute__((address_space(3))) v4i as3_v4i;

#if defined(__HIP_DEVICE_COMPILE__) && __has_builtin(__builtin_amdgcn_global_load_async_to_lds_b128)
#define HAVE_ASYNC 1
#else
#define HAVE_ASYNC 0
#if defined(__HIP_DEVICE_COMPILE__)
#warning "gfx1250 async-to-LDS builtin not found; using synchronous staging fallback"
#endif
#endif

#if defined(__HIP_DEVICE_COMPILE__) && __has_builtin(__builtin_amdgcn_s_wait_asynccnt)
#define WAIT_ASYNC(n) __builtin_amdgcn_s_wait_asynccnt(n)
#elif defined(__HIP_DEVICE_COMPILE__)
#define WAIT_ASYNC(n) asm volatile("s_wait_asynccnt " #n ::: "memory")
#else
#define WAIT_ASYNC(n) do {} while (0)
#endif

// ---------------------------------------------------------------------------

__global__ void zero_ws(uint32_t* __restrict__ p, int n) {
  for (int i = blockIdx.x * blockDim.x + threadIdx.x; i < n;
       i += gridDim.x * blockDim.x)
    p[i] = 0u;
}

// ---------------------------------------------------------------------------

__device__ __forceinline__ void emit_score(int a, float v, int ci,
                                           float* __restrict__ scores,
                                           uint32_t* __restrict__ classes,
                                           uint32_t* __restrict__ hist) {
  // sigmoid is monotonic: max over sigmoid(logits) == sigmoid(max logit)
  float s  = 1.0f / (1.0f + __expf(-v));
  float sv = (s > SCORE_TH) ? s : 0.0f;   // matches jnp.where(scores>T, scores, 0)
  scores[a]  = sv;
  classes[a] = (uint32_t)ci;
  atomicAdd(&hist[__float_as_uint(sv) >> 16], 1u);
}

// Issue exactly CHUNK_LOOPS async B128 copies per thread (dummy-padded so the
// per-wave s_wait_asynccnt immediate is uniform; async loads complete in order).
__device__ __forceinline__ void stage_async(const float* __restrict__ g0,
                                            float* __restrict__ lbuf) {
#if HAVE_ASYNC
#pragma unroll
  for (int k = 0; k < CHUNK_LOOPS; k++) {
    int t = (int)threadIdx.x + (k << 8);
    bool valid = (t < TILE_CH);
    const float* g = g0 + (valid ? (t << 2) : 0);
    float* l = lbuf + (valid ? (t << 2) : (TILE_CH << 2));   // dummy slot
    __builtin_amdgcn_global_load_async_to_lds_b128(
        (as1_v4i*)(uintptr_t)g,
        (as3_v4i*)(uint32_t)(uintptr_t)l,
        0 /*imm offset*/, 0 /*cpol*/);
  }
#else
  (void)g0; (void)lbuf;
#endif
}

__device__ __forceinline__ void stage_sync(const float* __restrict__ g0,
                                           float* __restrict__ lbuf) {
#pragma unroll
  for (int k = 0; k < CHUNK_LOOPS; k++) {
    int t = (int)threadIdx.x + (k << 8);
    if (t < TILE_CH) {
      float4 v = *reinterpret_cast<const float4*>(g0 + (t << 2));
      *reinterpret_cast<float4*>(lbuf + (t << 2)) = v;
    }
  }
}

// wave32-cooperative max/argmax over 90 classes: 3 strided LDS reads + 5 shfl_xor
__device__ __forceinline__ void reduce_tile(const float* __restrict__ buf,
                                            int tileStart, int lane, int wave,
                                            float* __restrict__ scores,
                                            uint32_t* __restrict__ classes,
                                            uint32_t* __restrict__ hist, int A) {
#pragma unroll
  for (int i = 0; i < TILE_A / 8; i++) {
    int aL = wave * (TILE_A / 8) + i;
    int a  = tileStart + aL;
    if (a >= A) break;
    const float* row = buf + aL * NUM_CLASSES;
    float v = -3.402823e38f;
    int   ci = 0x7fffffff;
    for (int j = lane; j < NUM_CLASSES; j += 32) {
      float x = row[j];
      if (x > v || (x == v && j < ci)) { v = x; ci = j; }
    }
#pragma unroll
    for (int off = 16; off; off >>= 1) {
      float v2 = __shfl_xor(v, off, 32);
      int   c2 = __shfl_xor(ci, off, 32);
      if (v2 > v || (v2 == v && c2 < ci)) { v = v2; ci = c2; }
    }
    if (lane == 0) emit_score(a, v, ci, scores, classes, hist);
  }
}

__global__ __launch_bounds__(THREADS) void score_kernel(
    const float* __restrict__ cls, float* __restrict__ scores,
    uint32_t* __restrict__ classes, uint32_t* __restrict__ hist,
    int numTiles, int A) {
  __shared__ __align__(16) float sCls[2][TILE_CH * 4 + 4];  // 2 x 23KB + dummy
  const int lane = threadIdx.x & 31;
  const int wave = threadIdx.x >> 5;

#if HAVE_ASYNC
  int cur  = 0;
  int tile = (int)blockIdx.x;
  if (tile < numTiles)
    stage_async(cls + (size_t)tile * TILE_A * NUM_CLASSES, sCls[0]);
  for (; tile < numTiles; tile += (int)gridDim.x) {
    int nxt = tile + (int)gridDim.x;
    if (nxt < numTiles) {
      stage_async(cls + (size_t)nxt * TILE_A * NUM_CLASSES, sCls[cur ^ 1]);
      WAIT_ASYNC(6);   // 6 newer asyncs in flight -> older tile fully resident
    } else {
      WAIT_ASYNC(0);
    }
    __syncthreads();
    reduce_tile(sCls[cur], tile * TILE_A, lane, wave, scores, classes, hist, A);
    __syncthreads();
    cur ^= 1;
  }
#else
  for (int tile = (int)blockIdx.x; tile < numTiles; tile += (int)gridDim.x) {
    stage_sync(cls + (size_t)tile * TILE_A * NUM_CLASSES, sCls[0]);
    __syncthreads();
    reduce_tile(sCls[0], tile * TILE_A, lane, wave, scores, classes, hist, A);
    __syncthreads();
  }
#endif
}

// Remainder anchors when A % TILE_A != 0 (not hit for A=196416, kept for safety)
__global__ void score_tail(const float* __restrict__ cls,
                           float* __restrict__ scores,
                           uint32_t* __restrict__ classes,
                           uint32_t* __restrict__ hist, int start, int A) {
  int a = start + blockIdx.x * blockDim.x + threadIdx.x;
  if (a >= A) return;
  float v = -3.402823e38f; int ci = 0;
  const float* row = cls + (size_t)a * NUM_CLASSES;
  for (int j = 0; j < NUM_CLASSES; j++) {
    float x = row[j];
    if (x > v) { v = x; ci = j; }
  }
  emit_score(a, v, ci, scores, classes, hist);
}

// ---------------------------------------------------------------------------
// Suffix scan of a 65536-bin histogram from the top; finds cutoff bin where the
// cumulative count crosses `target`. mode 0: coarse (target=100) -> ctrl[0]=bin,
// ctrl[1]=count strictly above. mode 1: fine (target=100-ctrl[1]) ->
// ctrl[2]=exact 32-bit threshold bit pattern.
__global__ __launch_bounds__(1024) void scan_hist(
    const uint32_t* __restrict__ hist, uint32_t* ctrl, int mode) {
  __shared__ uint32_t csum[1024];
  const int tid = threadIdx.x;
  uint32_t s = 0;
  const uint32_t* h = hist + tid * 64;
#pragma unroll 8
  for (int i = 0; i < 64; i++) s += h[i];
  csum[tid] = s;
  __syncthreads();
  if (tid == 0) {
    int target = (mode == 0) ? MAX_DET : (MAX_DET - (int)ctrl[1]);
    if (target < 1) target = 1;
    uint32_t cum = 0;
    int c;
    for (c = 1023; c > 0; c--) {
      if (cum + csum[c] >= (uint32_t)target) break;
      cum += csum[c];
    }
    const uint32_t* hc = hist + c * 64;
    int b;
    for (b = 63; b > 0; b--) {
      if (cum + hc[b] >= (uint32_t)target) break;
      cum += hc[b];
    }
    uint32_t bin = (uint32_t)(c * 64 + b);
    if (mode == 0) { ctrl[0] = bin; ctrl[1] = cum; }
    else           { ctrl[2] = (ctrl[0] << 16) | bin; }
  }
}

__global__ void fine_hist_k(const float* __restrict__ scores,
                            const uint32_t* __restrict__ ctrl,
                            uint32_t* __restrict__ fine, int A) {
  const uint32_t cb = ctrl[0];
  const int stride = gridDim.x * blockDim.x;
  for (int a = blockIdx.x * blockDim.x + threadIdx.x; a < A; a += stride) {
    __builtin_prefetch(&scores[a + stride], 0, 1);   // global_prefetch_b8
    uint32_t b = __float_as_uint(scores[a]);
    if ((b >> 16) == cb) atomicAdd(&fine[b & 0xFFFFu], 1u);
  }
}

__global__ void compact_k(const float* __restrict__ scores, uint32_t* ctrl,
                          float* __restrict__ candS, uint32_t* __restrict__ candI,
                          int A) {
  const uint32_t tb = ctrl[2];
  const int stride = gridDim.x * blockDim.x;
  for (int a = blockIdx.x * blockDim.x + threadIdx.x; a < A; a += stride) {
    uint32_t b = __float_as_uint(scores[a]);   // scores >= 0: bit order == value order
    if (b >= tb && b != 0u) {
      uint32_t pos = atomicAdd(&ctrl[3], 1u);
      if (pos < CAND_CAP) { candS[pos] = __uint_as_float(b); candI[pos] = (uint32_t)a; }
    }
  }
}

// ---------------------------------------------------------------------------
// Exact top-100 over the tiny candidate set; decode+clip boxes only for winners.
__global__ __launch_bounds__(256) void select_k(
    const float* __restrict__ candS, const uint32_t* __restrict__ candI,
    const uint32_t* __restrict__ ctrl, const float* __restrict__ anchors,
    const float* __restrict__ regression, const uint32_t* __restrict__ classes,
    float* __restrict__ out) {
  __shared__ float    ss[CAND_CAP];
  __shared__ uint32_t si[CAND_CAP];
  __shared__ float    rs[256];
  __shared__ uint32_t ra[256];
  __shared__ int      rp[256];

  const int tid = threadIdx.x;
  int nc = (int)ctrl[3];
  if (nc > CAND_CAP) nc = CAND_CAP;
  for (int i = tid; i < CAND_CAP; i += 256) {
    ss[i] = (i < nc) ? candS[i] : -1.0f;
    si[i] = (i < nc) ? candI[i] : 0xFFFFFFFFu;
  }
  __syncthreads();

  for (int k = 0; k < MAX_DET; k++) {
    float bv = -1.0f; uint32_t ba = 0xFFFFFFFFu; int bp = -1;
    for (int i = tid; i < CAND_CAP; i += 256) {
      float v = ss[i]; uint32_t a2 = si[i];
      if (v > bv || (v == bv && a2 < ba)) { bv = v; ba = a2; bp = i; }
    }
    rs[tid] = bv; ra[tid] = ba; rp[tid] = bp;
    __syncthreads();
#pragma unroll
    for (int off = 128; off; off >>= 1) {
      if (tid < off) {
        float v2 = rs[tid + off]; uint32_t a2 = ra[tid + off];
        if (v2 > rs[tid] || (v2 == rs[tid] && a2 < ra[tid])) {
          rs[tid] = v2; ra[tid] = a2; rp[tid] = rp[tid + off];
        }
      }
      __syncthreads();
    }
    if (tid == 0) {
      int p = rp[0]; float sv = rs[0]; uint32_t a = ra[0];
      float x1 = 0.f, y1 = 0.f, x2 = 0.f, y2 = 0.f, cw = 0.f, so = 0.f;
      if (p >= 0 && sv > 0.0f && a != 0xFFFFFFFFu) {
        float4 an = reinterpret_cast<const float4*>(anchors)[a];
        float4 rg = reinterpret_cast<const float4*>(regression)[a];
        float wa = an.z - an.x, ha = an.w - an.y;
        float cxa = an.x + 0.5f * wa, cya = an.y + 0.5f * ha;
        float cx = cxa + (rg.x * 0.1f) * wa;
        float cy = cya + (rg.y * 0.1f) * ha;
        float w = __expf(rg.z * 0.2f) * wa;
        float h = __expf(rg.w * 0.2f) * ha;
        x1 = fmaxf(cx - 0.5f * w, 0.0f);
        y1 = fmaxf(cy - 0.5f * h, 0.0f);
        x2 = fminf(cx + 0.5f * w, IMG_W);
        y2 = fminf(cy + 0.5f * h, IMG_H);
        cw = (float)classes[a];
        so = sv;
        ss[p] = -2.0f;   // consume winner
      }
      out[k * 4 + 0] = x1; out[k * 4 + 1] = y1;
      out[k * 4 + 2] = x2; out[k * 4 + 3] = y2;
      out[4 * MAX_DET + k] = so;        // scores block [400,500)
      out[5 * MAX_DET + k] = cw;        // classes block [500,600)
    }
    __syncthreads();
  }
}

// ---------------------------------------------------------------------------

extern "C" void kernel_launch(void* const* d_in, const int* in_sizes, int n_in,
                              void* d_out, int out_size, void* d_ws, size_t ws_size,
                              hipStream_t stream) {
  (void)n_in; (void)out_size; (void)ws_size;
  // inputs (setup_inputs order): x [unused, shape-only], regression, classification, anchors
  const float* regression     = (const float*)d_in[1];
  const float* classification = (const float*)d_in[2];
  const float* anchors        = (const float*)d_in[3];
  float* out = (float*)d_out;
  const int A = in_sizes[1] / 4;                 // 196416

  // workspace layout (u32 words)
  uint32_t* ws32   = (uint32_t*)d_ws;
  uint32_t* coarse = ws32;                       // 65536
  uint32_t* fineH  = ws32 + 65536;               // 65536
  uint32_t* ctrl   = ws32 + 131072;              // 16: [0]=coarseBin [1]=nAbove [2]=threshBits [3]=candCount
  float*    candS  = (float*)(ws32 + 131088);    // CAND_CAP
  uint32_t* candI  = ws32 + 131088 + CAND_CAP;   // CAND_CAP
  float*    scores = (float*)(ws32 + 131088 + 2 * CAND_CAP);        // A
  uint32_t* classW = ws32 + 131088 + 2 * CAND_CAP + (uint32_t)A;    // A

  zero_ws<<<256, 256, 0, stream>>>(ws32, 131088);

  const int tiles = A / TILE_A;
  if (tiles > 0) {
    int grid = tiles < 768 ? tiles : 768;
    score_kernel<<<grid, THREADS, 0, stream>>>(classification, scores, classW,
                                               coarse, tiles, A);
  }
  const int rem = A - tiles * TILE_A;
  if (rem > 0)
    score_tail<<<(rem + 255) / 256, 256, 0, stream>>>(classification, scores,
                                                      classW, coarse,
                                                      tiles * TILE_A, A);

  scan_hist<<<1, 1024, 0, stream>>>(coarse, ctrl, 0);
  fine_hist_k<<<512, 256, 0, stream>>>(scores, ctrl, fineH, A);
  scan_hist<<<1, 1024, 0, stream>>>(fineH, ctrl, 1);
  compact_k<<<512, 256, 0, stream>>>(scores, ctrl, candS, candI, A);
  select_k<<<1, 256, 0, stream>>>(candS, candI, ctrl, anchors, regression,
                                  classW, out);
}